// GroupSparseAE_68968584839792
// MI455X (gfx1250) — compile-verified
//
#include <hip/hip_runtime.h>
#include <cmath>

// ---------------------------------------------------------------------------
// GroupSparseAE on MI455X (gfx1250): FISTA loop = 30 x 3 x [512x2048,K=2048]
// FP32 GEMMs; all operands L2-resident (Gram 50MB << 192MB L2) -> compute
// bound on the FP32 WMMA pipe (v_wmma_f32_16x16x4_f32, reference precision).
// This round: double-buffered software pipeline with CDNA5 async-to-LDS
// copies (global_load_async_to_lds_b128 + s_wait_asynccnt) so stage k+1's
// global->LDS traffic overlaps stage k's 64 WMMAs; no VGPR staging.
// ---------------------------------------------------------------------------

typedef float v2f __attribute__((ext_vector_type(2)));
typedef float v4f __attribute__((ext_vector_type(4)));
typedef float v8f __attribute__((ext_vector_type(8)));

#define B_SZ     512
#define C_CH     3
#define N_IN     1024
#define D_DICT   2048
#define N_LAYERS 30
#define TAU_F    0.1f
#define LAM_F    0.1f

#define TM 128
#define TN 128
#define TK 32
#define LDSS 36      // padded row stride (floats): 16B-aligned rows, bank spread
#define LDSB_NN 132  // padded row stride for NN B tile [TK][TN]
#define NTHREADS 256
#define TILE_F (TM * LDSS)   // floats per (single) tile buffer

// ---- CDNA5 async copy: global -> LDS, one [128 rows][TK] NT tile ----------
// 4 x b128 per thread; tracked on ASYNCcnt, no VGPR staging.
__device__ __forceinline__ void async_tile_nt(float* ldsbuf, const float* gptr,
                                              int ld, int row0, int k0) {
  const int r  = threadIdx.x >> 3;        // 0..31
  const int c4 = (threadIdx.x & 7) << 2;  // 0,4,..,28
#pragma unroll
  for (int p = 0; p < 4; ++p) {
    const int row = r + p * 32;
    const float* g = gptr + (size_t)(row0 + row) * ld + k0 + c4;
    // low 32 bits of a flat shared pointer == wave-relative LDS byte address
    const unsigned lo = (unsigned)(unsigned long long)(const void*)&ldsbuf[row * LDSS + c4];
    asm volatile("global_load_async_to_lds_b128 %0, %1, off"
                 :: "v"(lo), "v"(g)
                 : "memory");
  }
}

__device__ __forceinline__ void wait_async0() {
  asm volatile("s_wait_asynccnt 0x0" ::: "memory");
}

// ---- sync loader (cold decoder path only) ---------------------------------
__device__ __forceinline__ void load_tile_nt(float* lds, const float* gptr,
                                             int ld, int row0, int k0) {
  const int r  = threadIdx.x >> 3;
  const int c4 = (threadIdx.x & 7) << 2;
#pragma unroll
  for (int p = 0; p < 4; ++p) {
    const int row = r + p * 32;
    v4f v = *(const v4f*)(gptr + (size_t)(row0 + row) * ld + k0 + c4);
    *(v4f*)&lds[row * LDSS + c4] = v;
  }
}

// ---- WMMA mainloop with async double buffer: C[128x128] += A * B^T --------
// lA/lB point at 2*TILE_F double buffers in LDS.
__device__ __forceinline__ void gemm_mainloop(const float* Ab, int lda,
                                              const float* Bb, int ldb, int K,
                                              int rowM0, int colN0,
                                              float* lA, float* lB,
                                              v8f acc[4][2]) {
  const int lane = threadIdx.x & 31;
  const int wave = threadIdx.x >> 5;
  const int wmB  = (wave >> 2) * 64;   // 2 waves tile M
  const int wnB  = (wave & 3) * 32;    // 4 waves tile N
  const int l15  = lane & 15;
  const int hi2  = (lane >> 4) << 1;   // lanes 16-31 hold K+2,K+3

  // prologue: stage 0 in flight
  async_tile_nt(lA, Ab, lda, rowM0, 0);
  async_tile_nt(lB, Bb, ldb, colN0, 0);

  for (int k0 = 0; k0 < K; k0 += TK) {
    const int par = (k0 / TK) & 1;
    float* cA = lA + par * TILE_F;
    float* cB = lB + par * TILE_F;

    wait_async0();      // my stage-s tile writes landed
    __syncthreads();    // everyone's landed; previous buffer now dead

    if (k0 + TK < K) {  // overlap stage s+1 copy with stage s compute
      async_tile_nt(lA + (par ^ 1) * TILE_F, Ab, lda, rowM0, k0 + TK);
      async_tile_nt(lB + (par ^ 1) * TILE_F, Bb, ldb, colN0, k0 + TK);
    }

#pragma unroll
    for (int kk = 0; kk < TK; kk += 4) {
      const int ko = kk + hi2;
      v2f af[4], bf[2];
#pragma unroll
      for (int i = 0; i < 4; ++i)
        af[i] = *(const v2f*)&cA[(wmB + i * 16 + l15) * LDSS + ko];
#pragma unroll
      for (int j = 0; j < 2; ++j)
        bf[j] = *(const v2f*)&cB[(wnB + j * 16 + l15) * LDSS + ko];
#pragma unroll
      for (int i = 0; i < 4; ++i)
#pragma unroll
        for (int j = 0; j < 2; ++j)
          acc[i][j] = __builtin_amdgcn_wmma_f32_16x16x4_f32(
              false, af[i], false, bf[j], (short)0, acc[i][j], false, false);
    }
  }
}

// ---- generic NT GEMM with plain store (Gram and precomp_y) ----------------
__global__ __launch_bounds__(NTHREADS)
void gemm_nt_kernel(const float* __restrict__ A, int lda, long long sA,
                    const float* __restrict__ Bm, int ldb, long long sB,
                    float* __restrict__ Cout, int ldc, long long sC, int K) {
  __shared__ float lA[2 * TILE_F];
  __shared__ float lB[2 * TILE_F];
  const int rowM0 = blockIdx.y * TM;
  const int colN0 = blockIdx.x * TN;
  const int z = blockIdx.z;
  v8f acc[4][2] = {};
  gemm_mainloop(A + (size_t)z * sA, lda, Bm + (size_t)z * sB, ldb, K,
                rowM0, colN0, lA, lB, acc);

  const int lane = threadIdx.x & 31;
  const int wave = threadIdx.x >> 5;
  const int wmB = (wave >> 2) * 64, wnB = (wave & 3) * 32;
  const int l15 = lane & 15;
  const int hi8 = (lane >> 4) << 3;  // lanes 16-31 hold rows M+8..M+15
  float* Cz = Cout + (size_t)z * sC;
#pragma unroll
  for (int i = 0; i < 4; ++i)
#pragma unroll
    for (int j = 0; j < 2; ++j) {
      const int col = colN0 + wnB + j * 16 + l15;
      const int mB  = rowM0 + wmB + i * 16 + hi8;
#pragma unroll
      for (int r = 0; r < 8; ++r)
        Cz[(size_t)(mB + r) * ldc + col] = acc[i][j][r];
    }
}

// ---- FISTA layer: grad GEMM fused with group soft-threshold + momentum ----
__global__ __launch_bounds__(NTHREADS)
void fista_kernel(const float* __restrict__ xtmp, const float* __restrict__ gram,
                  const float* __restrict__ y,    const float* __restrict__ xold,
                  float* __restrict__ xnew, float* __restrict__ xmom, float mu) {
  __shared__ float lA[2 * TILE_F];
  __shared__ float lB[2 * TILE_F];
  const int rowM0 = blockIdx.y * TM;
  const int colN0 = blockIdx.x * TN;
  const int z = blockIdx.z;
  const size_t slab = (size_t)B_SZ * D_DICT;
  const float* xt = xtmp + (size_t)z * slab;

  v8f acc[4][2] = {};
  // Gram is symmetric -> x*Gram == x*Gram^T: NT form, same fragment addressing.
  gemm_mainloop(xt, D_DICT, gram + (size_t)z * D_DICT * D_DICT, D_DICT, D_DICT,
                rowM0, colN0, lA, lB, acc);

  const int lane = threadIdx.x & 31;
  const int wave = threadIdx.x >> 5;
  const int wmB = (wave >> 2) * 64, wnB = (wave & 3) * 32;
  const int l15 = lane & 15;
  const int hi8 = (lane >> 4) << 3;
  const float* yz = y    + (size_t)z * slab;
  const float* xo = xold + (size_t)z * slab;
  float* xn_p = xnew + (size_t)z * slab;
  float* xm_p = xmom + (size_t)z * slab;
  const float lt = LAM_F * TAU_F;

#pragma unroll
  for (int i = 0; i < 4; ++i)
#pragma unroll
    for (int j = 0; j < 2; ++j) {
      const int col = colN0 + wnB + j * 16 + l15;  // 16-aligned base -> groups
      const int mB  = rowM0 + wmB + i * 16 + hi8;  // of 8 along D align to lanes
#pragma unroll
      for (int r = 0; r < 8; ++r) {
        const size_t idx = (size_t)(mB + r) * D_DICT + col;
        const float grad = acc[i][j][r] - yz[idx];
        const float v = xt[idx] - TAU_F * grad;
        // group-of-8 L2 norm along D == butterfly over 8 adjacent lanes
        float sq = v * v;
        sq += __shfl_xor(sq, 1, 32);
        sq += __shfl_xor(sq, 2, 32);
        sq += __shfl_xor(sq, 4, 32);
        const float nrm = sqrtf(sq);
        const float sc  = (nrm > lt) ? (1.0f - lt / nrm) : 0.0f;
        const float xnv = sc * fmaxf(v, 0.0f);     // one-sided shrink
        const float xov = xo[idx];
        xn_p[idx] = xnv;
        xm_p[idx] = fmaf(mu, xnv - xov, xnv);      // momentum for next layer
      }
    }
}

// ---- decoder: x_hat[b,c,n] = sum_d z[b,c,d] * W[c,d,n]  (NN GEMM, cold) ---
__global__ __launch_bounds__(NTHREADS)
void decoder_kernel(const float* __restrict__ zc, const float* __restrict__ W,
                    float* __restrict__ out) {
  __shared__ float lA[TILE_F];
  __shared__ float lB[TK * LDSB_NN];
  const int rowM0 = blockIdx.y * TM;
  const int colN0 = blockIdx.x * TN;
  const int c = blockIdx.z;
  const float* Az = zc + (size_t)c * B_SZ * D_DICT;  // [512][2048]
  const float* Wz = W  + (size_t)c * D_DICT * N_IN;  // [2048][1024]

  const int lane = threadIdx.x & 31;
  const int wave = threadIdx.x >> 5;
  const int wmB = (wave >> 2) * 64, wnB = (wave & 3) * 32;
  const int l15 = lane & 15;
  const int hi2 = (lane >> 4) << 1;
  v8f acc[4][2] = {};

  for (int k0 = 0; k0 < D_DICT; k0 += TK) {
    load_tile_nt(lA, Az, D_DICT, rowM0, k0);
    {  // B tile NN: rows k (TK=32) x cols n (TN=128), coalesced from W rows
      const int r  = threadIdx.x >> 5;          // 0..7
      const int c4 = (threadIdx.x & 31) << 2;   // 0..124
#pragma unroll
      for (int p = 0; p < 4; ++p) {
        const int row = r + p * 8;
        v4f v = *(const v4f*)(Wz + (size_t)(k0 + row) * N_IN + colN0 + c4);
        *(v4f*)&lB[row * LDSB_NN + c4] = v;
      }
    }
    __syncthreads();
#pragma unroll
    for (int kk = 0; kk < TK; kk += 4) {
      const int ko = kk + hi2;
      v2f af[4], bf[2];
#pragma unroll
      for (int i = 0; i < 4; ++i)
        af[i] = *(const v2f*)&lA[(wmB + i * 16 + l15) * LDSS + ko];
#pragma unroll
      for (int j = 0; j < 2; ++j) {
        bf[j][0] = lB[(ko + 0) * LDSB_NN + wnB + j * 16 + l15];
        bf[j][1] = lB[(ko + 1) * LDSB_NN + wnB + j * 16 + l15];
      }
#pragma unroll
      for (int i = 0; i < 4; ++i)
#pragma unroll
        for (int j = 0; j < 2; ++j)
          acc[i][j] = __builtin_amdgcn_wmma_f32_16x16x4_f32(
              false, af[i], false, bf[j], (short)0, acc[i][j], false, false);
    }
    __syncthreads();
  }

  const int hi8 = (lane >> 4) << 3;
  float* oz = out + (size_t)c * N_IN;  // out is [B][C][N]
#pragma unroll
  for (int i = 0; i < 4; ++i)
#pragma unroll
    for (int j = 0; j < 2; ++j) {
      const int col = colN0 + wnB + j * 16 + l15;
      const int mB  = rowM0 + wmB + i * 16 + hi8;
#pragma unroll
      for (int r = 0; r < 8; ++r)
        oz[(size_t)(mB + r) * (C_CH * N_IN) + col] = acc[i][j][r];
    }
}

// ---------------------------------------------------------------------------
extern "C" void kernel_launch(void* const* d_in, const int* in_sizes, int n_in,
                              void* d_out, int out_size, void* d_ws, size_t ws_size,
                              hipStream_t stream) {
  const float* x = (const float*)d_in[0];   // [B][C][N]
  const float* W = (const float*)d_in[1];   // [C][D][N]
  float* out = (float*)d_out;               // [B][C][N]
  float* ws  = (float*)d_ws;

  const size_t slab = (size_t)B_SZ * D_DICT;           // per-channel state
  float* gram = ws;                                    // 3 * D * D   (50.3 MB)
  float* yb   = gram + (size_t)C_CH * D_DICT * D_DICT; // 3 * slab    (12.6 MB)
  float* s0   = yb + (size_t)C_CH * slab;              // ping: old|tmp
  float* s1   = s0 + 2 * (size_t)C_CH * slab;          // pong: old|tmp

  // FISTA starts from zeros (graph-capturable memset node)
  hipMemsetAsync(s0, 0, 2 * (size_t)C_CH * slab * sizeof(float), stream);

  const dim3 blk(NTHREADS);

  // precomp_W[c] = W_c @ W_c^T   (M=N=2048, K=1024)
  gemm_nt_kernel<<<dim3(D_DICT / TN, D_DICT / TM, C_CH), blk, 0, stream>>>(
      W, N_IN, (long long)D_DICT * N_IN,
      W, N_IN, (long long)D_DICT * N_IN,
      gram, D_DICT, (long long)D_DICT * D_DICT, N_IN);

  // precomp_y[c] = x_c @ W_c^T   (M=512, N=2048, K=1024); x row stride = C*N
  gemm_nt_kernel<<<dim3(D_DICT / TN, B_SZ / TM, C_CH), blk, 0, stream>>>(
      x, C_CH * N_IN, (long long)N_IN,
      W, N_IN, (long long)D_DICT * N_IN,
      yb, D_DICT, (long long)slab, N_IN);

  // 30 FISTA layers; t-sequence is data-independent -> precompute mu on host
  float t = 1.0f;
  float* ro = s0;                        // x_old
  float* rt = s0 + (size_t)C_CH * slab;  // x_tmp
  float* wo = s1;
  float* wt = s1 + (size_t)C_CH * slab;
  for (int it = 0; it < N_LAYERS; ++it) {
    const float tn = 0.5f * (1.0f + sqrtf(1.0f + 4.0f * t * t));
    const float mu = (t - 1.0f) / tn;
    fista_kernel<<<dim3(D_DICT / TN, B_SZ / TM, C_CH), blk, 0, stream>>>(
        rt, gram, yb, ro, wo, wt, mu);
    t = tn;
    float* sw;
    sw = ro; ro = wo; wo = sw;   // x_old <- x_new
    sw = rt; rt = wt; wt = sw;   // x_tmp <- x_mom
  }

  // decoder: x_hat = z @ W   (z = final x_new, now in ro)
  decoder_kernel<<<dim3(N_IN / TN, B_SZ / TM, C_CH), blk, 0, stream>>>(ro, W, out);
}